// BaseNeuron_34600256537582
// MI455X (gfx1250) — compile-verified
//
#include <hip/hip_runtime.h>

// LIF spiking-neuron forward scan, MI455X (gfx1250, wave32).
// Memory-bound: 105 MB read + 105 MB write => ~9 us floor at 23.3 TB/s.
// One thread owns 4 consecutive 'd' channels of one batch row, walks T=100.
// Recurrence: mem = spike_prev ? x_t : mem*sigmoid(decay) + x_t;
//             spike = (mem > 0.5).

typedef __attribute__((ext_vector_type(4))) float v4f;

__global__ __launch_bounds__(128) void lif_fwd_kernel(
    const float* __restrict__ xg,      // (B, T, D) fp32
    const float* __restrict__ decay,   // (1,) fp32
    float* __restrict__ outg)          // (B, T, D) fp32 spikes
{
  constexpr int T  = 100;
  constexpr int D  = 8192;
  constexpr int NV = D / 4;            // 2048 float4 per (b,t) row (power of 2)
  constexpr float THRESH = 0.5f;

  const int tid = blockIdx.x * blockDim.x + threadIdx.x;  // 0 .. B*NV-1
  const int b   = tid >> 11;           // tid / NV
  const int v   = tid & (NV - 1);      // tid % NV

  // sigmoid(decay) — uniform across the grid; decay is a device scalar.
  const float dk = 1.0f / (1.0f + __expf(-decay[0]));

  const v4f* __restrict__ xp = reinterpret_cast<const v4f*>(xg) +
                               (size_t)b * T * NV + v;
  v4f* __restrict__ op = reinterpret_cast<v4f*>(outg) +
                         (size_t)b * T * NV + v;

  // t = 0: mem = x0, spike = (mem > THRESH)
  v4f m = __builtin_nontemporal_load(xp);
  v4f s;
#pragma unroll
  for (int c = 0; c < 4; ++c) s[c] = (m[c] > THRESH) ? 1.0f : 0.0f;
  __builtin_nontemporal_store(s, op);

  // Steady state: t = 1 .. T-3, branchless body with prefetch 2 steps ahead
  // (32 KB stride per thread) — gfx1250 global_prefetch_b8.
  for (int t = 1; t < T - 2; ++t) {
    xp += NV;
    op += NV;
    __builtin_prefetch((const void*)(xp + 2 * NV), 0, 3);

    v4f xt = __builtin_nontemporal_load(xp);
#pragma unroll
    for (int c = 0; c < 4; ++c) {
      const float carried = (s[c] != 0.0f) ? 0.0f : m[c] * dk;
      m[c] = carried + xt[c];
      s[c] = (m[c] > THRESH) ? 1.0f : 0.0f;
    }
    __builtin_nontemporal_store(s, op);
  }

  // Epilogue: last two timesteps, no prefetch.
#pragma unroll
  for (int t = T - 2; t < T; ++t) {
    xp += NV;
    op += NV;
    v4f xt = __builtin_nontemporal_load(xp);
#pragma unroll
    for (int c = 0; c < 4; ++c) {
      const float carried = (s[c] != 0.0f) ? 0.0f : m[c] * dk;
      m[c] = carried + xt[c];
      s[c] = (m[c] > THRESH) ? 1.0f : 0.0f;
    }
    __builtin_nontemporal_store(s, op);
  }
}

extern "C" void kernel_launch(void* const* d_in, const int* in_sizes, int n_in,
                              void* d_out, int out_size, void* d_ws, size_t ws_size,
                              hipStream_t stream) {
  const float* x     = (const float*)d_in[0];   // (32, 100, 8192) fp32
  const float* decay = (const float*)d_in[1];   // (1,) fp32
  float* out         = (float*)d_out;           // (32, 100, 8192) fp32

  constexpr int B  = 32;
  constexpr int D  = 8192;
  constexpr int NV = D / 4;
  const int total_threads = B * NV;             // 65536
  const int block = 128;                        // 4 wave32 per block
  const int grid  = total_threads / block;      // 512 blocks: finer WGP balance

  lif_fwd_kernel<<<grid, block, 0, stream>>>(x, decay, out);
}